// GCNBlock_7945689497773
// MI455X (gfx1250) — compile-verified
//
#include <hip/hip_runtime.h>
#include <math.h>
#include <stdint.h>

typedef __attribute__((ext_vector_type(2))) float v2f;
typedef __attribute__((ext_vector_type(4))) float v4f;
typedef __attribute__((ext_vector_type(8))) float v8f;

#define N_NODES 10000
#define N_EDGES 160000
#define DIM     1024
#define MT      32        // rows (M) per block in the GEMM

// ---------------------------------------------------------------- degrees ---
__global__ __launch_bounds__(256) void k_deg_init(float* __restrict__ deg) {
    int n = blockIdx.x * 256 + threadIdx.x;
    if (n < N_NODES) deg[n] = 1.0f;   // self loop
}

__global__ __launch_bounds__(256) void k_deg_edges(const int* __restrict__ ei,
                                                   float* __restrict__ deg) {
    int e = blockIdx.x * 256 + threadIdx.x;
    if (e < N_EDGES) {
        int d = ei[N_EDGES + e];      // edge_index[1] = dst
        atomicAdd(&deg[d], 1.0f);
    }
}

__global__ __launch_bounds__(256) void k_dinv(const float* __restrict__ deg,
                                              float* __restrict__ dinv) {
    int n = blockIdx.x * 256 + threadIdx.x;
    if (n < N_NODES) {
        float dg = deg[n];
        dinv[n] = dg > 0.0f ? rsqrtf(dg) : 0.0f;
    }
}

// --------------------------------------------------------- GEMM  h = x @ W ---
// 313 blocks x 32 rows. 8 waves/block; wave w computes rows[32] x cols[w*128,+128).
// A-tile staged into LDS with GLOBAL_LOAD_ASYNC_TO_LDS_B128 (ASYNCcnt path);
// math on the full-precision fp32 matrix pipe: v_wmma_f32_16x16x4_f32.
// Each B frag (2 dword loads) feeds TWO WMMAs (rows 0-15 and rows 16-31).
__global__ __launch_bounds__(256) void k_gemm_wmma(const float* __restrict__ x,
                                                   const float* __restrict__ W,
                                                   float* __restrict__ h) {
    __shared__ __align__(16) float As[MT * DIM];   // 128 KB A-tile
    const int m0  = blockIdx.x * MT;
    const int tid = threadIdx.x;
    const int rows = (N_NODES - m0) < MT ? (N_NODES - m0) : MT;   // 32, or 16 in last block

    // --- async stage: global -> LDS, 16B per lane per issue, no VGPR round-trip
    {
        const unsigned lds0  = (unsigned)(uintptr_t)&As[0];   // wave-relative LDS offset
        const char*    gp    = (const char*)(x + (size_t)m0 * DIM);
        const int      total = rows * DIM * 4;                // bytes (multiple of 4096)
        for (int off = tid * 16; off < total; off += 256 * 16) {
            unsigned laddr = lds0 + (unsigned)off;
            uint64_t gaddr = (uint64_t)(uintptr_t)(gp + off);
            asm volatile("global_load_async_to_lds_b128 %0, %1, off"
                         :: "v"(laddr), "v"(gaddr) : "memory");
        }
        // zero-pad tail rows of the tile (only the last block does work here)
        for (int off = tid * 16 + rows * DIM * 4; off < MT * DIM * 4; off += 256 * 16) {
            *(v4f*)((char*)&As[0] + off) = (v4f){0.f, 0.f, 0.f, 0.f};
        }
        asm volatile("s_wait_asynccnt 0x0" ::: "memory");
    }
    __syncthreads();

    const int lane = tid & 31;
    const int wave = tid >> 5;
    const int nl   = lane & 15;        // N (B/C) or M (A) within 16
    const int kh   = lane >> 4;        // K-half: lanes 16-31 hold K=2,3
    const int nb   = wave * 128;       // column base of this wave's strip

    v8f acc0[8], acc1[8];
    #pragma unroll
    for (int j = 0; j < 8; ++j) { acc0[j] = {}; acc1[j] = {}; }

    for (int k0 = 0; k0 < DIM; k0 += 4) {
        const int ka = k0 + kh * 2;
        // A frags 16x4: lane = M, VGPR v = K  (one ds_load_b64 each)
        v2f a0 = *(const v2f*)(&As[nl * DIM + ka]);          // rows m0+0  .. m0+15
        v2f a1 = *(const v2f*)(&As[(16 + nl) * DIM + ka]);   // rows m0+16 .. m0+31

        const float* w0 = W + (size_t)ka * DIM + nb + nl;    // row K=ka
        const float* w1 = w0 + DIM;                          // row K=ka+1
        #pragma unroll
        for (int j = 0; j < 8; ++j) {
            // B frag 4x16: lane = N, VGPR v = K — reused by both M-tiles
            v2f b;
            b.x = w0[j * 16];
            b.y = w1[j * 16];
            acc0[j] = __builtin_amdgcn_wmma_f32_16x16x4_f32(
                false, a0, false, b, (short)0, acc0[j], false, false);
            acc1[j] = __builtin_amdgcn_wmma_f32_16x16x4_f32(
                false, a1, false, b, (short)0, acc1[j], false, false);
        }
    }

    // C/D layout: VGPR v holds M = v + kh*8, lanes give N.
    float* hp = h + nb + nl;
    #pragma unroll
    for (int j = 0; j < 8; ++j) {
        #pragma unroll
        for (int v = 0; v < 8; ++v) {
            const int r0 = m0 + kh * 8 + v;        // always < N_NODES (10000 % 32 == 16)
            const int r1 = r0 + 16;
            hp[(size_t)r0 * DIM + j * 16] = acc0[j][v];
            if (r1 < N_NODES) hp[(size_t)r1 * DIM + j * 16] = acc1[j][v];
        }
    }
}

// ---------------------------------------- self-loop message + bias (init) ---
__global__ __launch_bounds__(256) void k_self_bias(const float* __restrict__ h,
                                                   const float* __restrict__ dinv,
                                                   const float* __restrict__ bias,
                                                   float* __restrict__ out) {
    const int n = blockIdx.x;
    const int t = threadIdx.x;
    const float di = dinv[n];
    const float c  = di * di;                      // self-loop norm
    const v4f hv = *(const v4f*)(h + (size_t)n * DIM + t * 4);
    const v4f bv = *(const v4f*)(bias + t * 4);
    v4f o;
    o.x = hv.x * c + bv.x;
    o.y = hv.y * c + bv.y;
    o.z = hv.z * c + bv.z;
    o.w = hv.w * c + bv.w;
    *(v4f*)(out + (size_t)n * DIM + t * 4) = o;
}

// --------------------------------------------- edge gather + scatter-add ----
__global__ __launch_bounds__(256) void k_edge_scatter(const float* __restrict__ h,
                                                      const int* __restrict__ ei,
                                                      const float* __restrict__ dinv,
                                                      float* __restrict__ out) {
    const int e = blockIdx.x;
    const int s = ei[e];                 // src
    const int d = ei[N_EDGES + e];       // dst
    const float c = dinv[s] * dinv[d];
    const int t = threadIdx.x;
    const v4f hv = *(const v4f*)(h + (size_t)s * DIM + t * 4);
    float* op = out + (size_t)d * DIM + t * 4;
    atomicAdd(op + 0, hv.x * c);
    atomicAdd(op + 1, hv.y * c);
    atomicAdd(op + 2, hv.z * c);
    atomicAdd(op + 3, hv.w * c);
}

// ------------------------------------------------------------------ ReLU ----
__global__ __launch_bounds__(256) void k_relu(float* __restrict__ out) {
    size_t i = (size_t)blockIdx.x * 256 + threadIdx.x;
    float v = out[i];
    out[i] = v > 0.0f ? v : 0.0f;
}

// -------------------------------------------------------------- launcher ----
extern "C" void kernel_launch(void* const* d_in, const int* in_sizes, int n_in,
                              void* d_out, int out_size, void* d_ws, size_t ws_size,
                              hipStream_t stream) {
    const float* x  = (const float*)d_in[0];   // [N, D]
    const int*   ei = (const int*)d_in[1];     // [2, E]
    const float* W  = (const float*)d_in[2];   // [D, D]
    const float* b  = (const float*)d_in[3];   // [D]
    float* out = (float*)d_out;                // [N, D]

    float* ws   = (float*)d_ws;
    float* deg  = ws;                          // N_NODES floats
    float* dinv = ws + 10016;                  // N_NODES floats
    float* h    = ws + 32768;                  // N_NODES*DIM floats (40 MB)

    k_deg_init    <<<(N_NODES + 255) / 256, 256, 0, stream>>>(deg);
    k_deg_edges   <<<(N_EDGES + 255) / 256, 256, 0, stream>>>(ei, deg);
    k_dinv        <<<(N_NODES + 255) / 256, 256, 0, stream>>>(deg, dinv);
    k_gemm_wmma   <<<(N_NODES + MT - 1) / MT, 256, 0, stream>>>(x, W, h);
    k_self_bias   <<<N_NODES,               256, 0, stream>>>(h, dinv, b, out);
    k_edge_scatter<<<N_EDGES,               256, 0, stream>>>(h, ei, dinv, out);
    k_relu        <<<(N_NODES * DIM) / 256, 256, 0, stream>>>(out);
}